// MME2E_Sparse_74371653697907
// MI455X (gfx1250) — compile-verified
//
#include <hip/hip_runtime.h>
#include <stdint.h>

// ---------------------------------------------------------------------------
// Sparse-by-CDF attention masking (MME2E):
//   attn  [N, 144]  -> softmax -> top-p(0.9) boolean mask (stable-sort rule)
//   out   [N, 64, 144] = features * mask  (mask broadcast over channels)
//
// Bandwidth-bound: ~1.21 GB HBM traffic -> ~52 us at 23.3 TB/s. No matrix op
// exists in this workload (elementwise masking), so WMMA does not apply; the
// CDNA5-specific path used is the async global->LDS DMA (ASYNCcnt) for the
// attention row, plus B128 vector memory ops and non-temporal hints on the
// once-streamed feature data (working set 1.2 GB >> 192 MB L2).
//
// This revision fully unrolls the C==64 streaming loop: 9 NT b128 loads are
// issued back-to-back per wave (9x memory-level parallelism vs the rolled
// loop), and the %144 per-iteration modulo is strength-reduced to an
// add/compare (step 1024 mod 144 == 16).
// ---------------------------------------------------------------------------

typedef float v4f __attribute__((ext_vector_type(4)));

#define L        144      // H*W = 12*12
#define THREADS  256      // 8 waves (wave32)
#define CDF_T    0.9f

__device__ __forceinline__ void compute_mask_stage(
    const float* __restrict__ attn, float* s_vals, float* s_mask, float* s_red,
    int n, int tid)
{
    // ---- Stage 1: async-copy the 144-float attention row into LDS --------
    // 36 lanes each move one b128 via the CDNA5 async-DMA path (ASYNCcnt).
    {
        const float* attn_row = attn + (size_t)n * L;   // uniform -> SGPR pair
        if (tid < L / 4) {
            // addrspace(3) offsets live in the low 32 bits of the flat address
            unsigned lds_off = (unsigned)(uintptr_t)(&s_vals[0]) + (unsigned)tid * 16u;
            unsigned voff    = (unsigned)tid * 16u;
            asm volatile("global_load_async_to_lds_b128 %0, %1, %2"
                         :: "v"(lds_off), "v"(voff), "s"(attn_row)
                         : "memory");
        }
#if __has_builtin(__builtin_amdgcn_s_wait_asynccnt)
        __builtin_amdgcn_s_wait_asynccnt(0);
#else
        asm volatile("s_wait_asynccnt 0" ::: "memory");
#endif
    }
    __syncthreads();

    // ---- Stage 2: row max (wave 0 reduces) --------------------------------
    if (tid < 32) {
        float m = -3.402823466e38f;
        for (int j = tid; j < L; j += 32) m = fmaxf(m, s_vals[j]);
        #pragma unroll
        for (int d = 16; d >= 1; d >>= 1) m = fmaxf(m, __shfl_xor(m, d, 32));
        if (tid == 0) s_red[0] = m;
    }
    __syncthreads();
    const float mx = s_red[0];

    // ---- Stage 3: unnormalized softmax weights e_i = exp(x_i - max) -------
    if (tid < L) s_vals[tid] = __expf(s_vals[tid] - mx);
    __syncthreads();

    // ---- Stage 4: row sum -------------------------------------------------
    if (tid < 32) {
        float s = 0.0f;
        for (int j = tid; j < L; j += 32) s += s_vals[j];
        #pragma unroll
        for (int d = 16; d >= 1; d >>= 1) s += __shfl_xor(s, d, 32);
        if (tid == 0) s_red[1] = s;
    }
    __syncthreads();
    const float thresh = CDF_T * s_red[1];   // compare unnormalized: S < 0.9*sum

    // ---- Stage 5: top-p mask without sorting ------------------------------
    // Keep i iff the mass strictly ahead of i in stable descending order is
    // below the threshold (exactly reproduces cumsum<cdf plus the k-th entry,
    // including the stable tie-break of jnp.argsort(-t)).
    // All 144 lanes read the same s_vals[j] each iteration -> LDS broadcast.
    if (tid < L) {
        const float pi = s_vals[tid];
        float S = 0.0f;
        for (int j = 0; j < L; ++j) {
            const float pj = s_vals[j];
            if (pj > pi || (pj == pi && j < tid)) S += pj;
        }
        s_mask[tid] = (S < thresh) ? 1.0f : 0.0f;
    }
    __syncthreads();
}

// Specialized: C known at compile time and C*L/4 divisible by THREADS.
template <int C_T>
__global__ __launch_bounds__(THREADS) void MME2E_sparse_mask_mul_fixed(
    const float* __restrict__ features,   // [N, C_T, L]
    const float* __restrict__ attn,       // [N, L]
    float* __restrict__ out)              // [N, C_T, L]
{
    __shared__ __align__(16) float s_vals[L];
    __shared__ __align__(16) float s_mask[L];
    __shared__ float s_red[2];

    const int n   = blockIdx.x;
    const int tid = threadIdx.x;

    compute_mask_stage(attn, s_vals, s_mask, s_red, n, tid);

    // ---- Stage 6: stream features * mask (fully unrolled) -----------------
    constexpr int NF4  = (C_T * L) / 4;        // float4s per sample (2304)
    constexpr int ITER = NF4 / THREADS;        // 9
    static_assert(NF4 % THREADS == 0, "specialized path requires exact tiling");
    constexpr int STEP = (THREADS * 4) % L;    // 16: l-index advance per trip

    const size_t base = (size_t)n * (size_t)(C_T * L);
    const v4f* __restrict__ src = (const v4f*)(features + base);
    v4f*       __restrict__ dst = (v4f*)(out + base);

    v4f v[ITER];
    int lidx[ITER];
    int l4 = (tid * 4) % L;                    // 0..140, step 4 within a row

    #pragma unroll
    for (int k = 0; k < ITER; ++k) {           // 9 NT b128 loads in flight
        lidx[k] = l4;
        v[k] = __builtin_nontemporal_load(&src[tid + k * THREADS]);
        l4 += STEP;
        if (l4 >= L) l4 -= L;
    }
    #pragma unroll
    for (int k = 0; k < ITER; ++k) {
        v4f m = *(const v4f*)(&s_mask[lidx[k]]);           // ds_load_b128
        __builtin_nontemporal_store(v[k] * m, &dst[tid + k * THREADS]);
    }
}

// Generic fallback for unexpected shapes.
__global__ __launch_bounds__(THREADS) void MME2E_sparse_mask_mul_generic(
    const float* __restrict__ features,
    const float* __restrict__ attn,
    float* __restrict__ out,
    int C)
{
    __shared__ __align__(16) float s_vals[L];
    __shared__ __align__(16) float s_mask[L];
    __shared__ float s_red[2];

    const int n   = blockIdx.x;
    const int tid = threadIdx.x;

    compute_mask_stage(attn, s_vals, s_mask, s_red, n, tid);

    const size_t base = (size_t)n * (size_t)C * (size_t)L;
    const v4f* __restrict__ src = (const v4f*)(features + base);
    v4f*       __restrict__ dst = (v4f*)(out + base);
    const int nf4 = (C * L) / 4;

    for (int f4 = tid; f4 < nf4; f4 += THREADS) {
        const int l4 = (f4 * 4) % L;
        v4f v = __builtin_nontemporal_load(&src[f4]);
        v4f m = *(const v4f*)(&s_mask[l4]);
        __builtin_nontemporal_store(v * m, &dst[f4]);
    }
}

extern "C" void kernel_launch(void* const* d_in, const int* in_sizes, int n_in,
                              void* d_out, int out_size, void* d_ws, size_t ws_size,
                              hipStream_t stream) {
    const float* features = (const float*)d_in[0];   // [N, C, H, W] fp32
    const float* attn     = (const float*)d_in[1];   // [N, H, W]    fp32
    float*       out      = (float*)d_out;           // [N, C, H, W] fp32

    const int NL = in_sizes[1];        // N * 144
    const int N  = NL / L;             // 16384
    const int C  = in_sizes[0] / NL;   // 64

    if (C == 64) {
        MME2E_sparse_mask_mul_fixed<64><<<N, THREADS, 0, stream>>>(features, attn, out);
    } else {
        MME2E_sparse_mask_mul_generic<<<N, THREADS, 0, stream>>>(features, attn, out, C);
    }
}